// Yolo_v3_loss_78778290143661
// MI455X (gfx1250) — compile-verified
//
#include <hip/hip_runtime.h>
#include <math.h>

// ---------------- problem constants (fixed by the reference) ----------------
#define NB 32
#define NA 3
#define NS 52
#define CH 85                 // 5 + 80 prediction channels
#define NCLS 80
#define TCH 6
#define NCELLS (NB * NA * NS * NS)       // 259584
#define BDIM 128                          // 4 waves of 32
#define WAVES 4
#define CELLS_PER_WAVE 32
#define F4_PER_WAVE ((CELLS_PER_WAVE * CH) / 4)   // 680 float4 per wave chunk
#define NBLOCKS (NCELLS / BDIM)           // 2028, exact (no tail)

// ---------------- CDNA5 feature detection ----------------
#if defined(__has_builtin)
#  if __has_builtin(__builtin_amdgcn_global_load_async_to_lds_b128)
#    define USE_ASYNC_LDS 1
#  endif
#  if __has_builtin(__builtin_amdgcn_wmma_f32_16x16x4_f32)
#    define USE_WMMA 1
#  endif
#endif

#define GLOBAL_AS __attribute__((address_space(1)))
#define LDS_AS    __attribute__((address_space(3)))

typedef int v4i __attribute__((vector_size(16)));       // matches builtin param
typedef __attribute__((ext_vector_type(2))) float v2f;
typedef __attribute__((ext_vector_type(8))) float v8f;

// ---------------- helpers ----------------
__device__ __forceinline__ float wave_sum32(float v) {
  #pragma unroll
  for (int off = 16; off > 0; off >>= 1) v += __shfl_xor(v, off, 32);
  return v;
}

// Reduce two f32 accumulators across the full wave32 with one
// V_WMMA_F32_16X16X4_F32 (exact f32 FMA math) and atomically add the totals.
// B is a 2-column selector: col0 picks K{0,2} (va of lanes m and m+16),
// col1 picks K{1,3} (vb). After summing the 8 D VGPRs, lanes 0/16 hold the
// two halves of sum(va) and lanes 1/17 the halves of sum(vb) -> those four
// lanes atomicAdd directly; no cross-lane shuffles needed at all.
__device__ __forceinline__ void wave_reduce_pair_atomic(float va, float vb,
                                                        int lane, float* wsa,
                                                        float* wsb) {
#if defined(USE_WMMA)
  v2f a; a[0] = va; a[1] = vb;
  v2f b;
  b[0] = ((lane & 15) == 0) ? 1.0f : 0.0f;   // B[0][0] (lanes 0-15) / B[2][0] (16-31)
  b[1] = ((lane & 15) == 1) ? 1.0f : 0.0f;   // B[1][1]              / B[3][1]
  v8f c = {};
  c = __builtin_amdgcn_wmma_f32_16x16x4_f32(false, a, false, b, (short)0, c,
                                            false, false);
  const float s =
      ((c[0] + c[1]) + (c[2] + c[3])) + ((c[4] + c[5]) + (c[6] + c[7]));
  const int col = lane & 15;
  if (col == 0) atomicAdd(wsa, s);         // lanes 0 and 16: column-0 halves
  if (col == 1) atomicAdd(wsb, s);         // lanes 1 and 17: column-1 halves
#else
  const float ta = wave_sum32(va);
  const float tb = wave_sum32(vb);
  if (lane == 0) {
    atomicAdd(wsa, ta);
    atomicAdd(wsb, tb);
  }
#endif
}

__device__ __forceinline__ float bce_pos_term(float x) {
  // max(x,0) + log1p(exp(-|x|))
  return fmaxf(x, 0.0f) + log1pf(expf(-fabsf(x)));
}

// ---------------- kernels ----------------
__global__ void yolo_zero_ws(float* ws) {
  if (threadIdx.x < 6) ws[threadIdx.x] = 0.0f;
}

__global__ __launch_bounds__(BDIM) void yolo_main(
    const float* __restrict__ pred, const float* __restrict__ tgt,
    const float* __restrict__ anchors, float* __restrict__ ws) {
  __shared__ float smem[WAVES * CELLS_PER_WAVE * CH];  // 43,520 B

  const int tid  = threadIdx.x;
  const int lane = tid & 31;
  const int wv   = tid >> 5;
  const int cell = blockIdx.x * BDIM + tid;            // exact grid, no tail

  // ---- stage this wave's 32 cells (2720 floats, contiguous) into LDS ----
  const int chunk_cell = blockIdx.x * BDIM + wv * CELLS_PER_WAVE;
  const float4* gp = (const float4*)(pred + (size_t)chunk_cell * CH);
  float4* sp = ((float4*)smem) + (size_t)wv * F4_PER_WAVE;

  for (int j = lane; j < F4_PER_WAVE; j += 32) {
#if defined(USE_ASYNC_LDS)
    // AS1 pointer: flat and global addresses coincide numerically.
    // AS3 pointer: per ISA, LDS offset == low 32 bits of the generic address.
    __builtin_amdgcn_global_load_async_to_lds_b128(
        (GLOBAL_AS v4i*)(size_t)(gp + j),
        (LDS_AS v4i*)(unsigned)(size_t)(sp + j), 0, 0);
#else
    sp[j] = gp[j];
#endif
  }
#if defined(USE_ASYNC_LDS)
#  if defined(__has_builtin) && __has_builtin(__builtin_amdgcn_s_wait_asynccnt)
  __builtin_amdgcn_s_wait_asynccnt(0);
#  else
  asm volatile("s_wait_asynccnt 0x0" ::: "memory");
#  endif
#endif
  __syncthreads();

  // lane's cell data: smem + tid*85 (stride 85 floats across lanes ->
  // banks 21*lane mod 64: conflict-free)
  const float* my = smem + (size_t)tid * CH;

  float acc_noobj = 0.f, acc_obj = 0.f, acc_box = 0.f, acc_cls = 0.f;
  float cnt_obj = 0.f, cnt_noobj = 0.f;

  const float t0 = tgt[(size_t)cell * TCH + 0];
  const float p0 = my[0];

  if (t0 == 0.0f) {
    cnt_noobj = 1.0f;
    acc_noobj = bce_pos_term(p0);                 // z = 0
  } else if (t0 == 1.0f) {
    cnt_obj = 1.0f;
    const int a = (cell / (NS * NS)) % NA;
    const float aw = anchors[a * 2 + 0];
    const float ah = anchors[a * 2 + 1];
    const float tx = tgt[(size_t)cell * TCH + 1];
    const float ty = tgt[(size_t)cell * TCH + 2];
    const float tw = tgt[(size_t)cell * TCH + 3];
    const float th = tgt[(size_t)cell * TCH + 4];
    const float p1 = my[1], p2 = my[2], p3 = my[3], p4 = my[4];

    const float sx = 1.0f / (1.0f + expf(-p1));
    const float sy = 1.0f / (1.0f + expf(-p2));
    const float pw = expf(p3) * aw;
    const float ph = expf(p4) * ah;

    // midpoint IOU (detached in reference; forward-only here)
    const float ax1 = sx - pw * 0.5f, ay1 = sy - ph * 0.5f;
    const float ax2 = sx + pw * 0.5f, ay2 = sy + ph * 0.5f;
    const float bx1 = tx - tw * 0.5f, by1 = ty - th * 0.5f;
    const float bx2 = tx + tw * 0.5f, by2 = ty + th * 0.5f;
    const float iw = fmaxf(fminf(ax2, bx2) - fmaxf(ax1, bx1), 0.0f);
    const float ih = fmaxf(fminf(ay2, by2) - fmaxf(ay1, by1), 0.0f);
    const float inter = iw * ih;
    const float area_a = fabsf((ax2 - ax1) * (ay2 - ay1));
    const float area_b = fabsf((bx2 - bx1) * (by2 - by1));
    const float iou = inter / (area_a + area_b - inter + 1e-6f);

    // object BCE: logits p0 vs z = iou (t0 == 1)
    acc_obj = bce_pos_term(p0) - p0 * iou;

    // box MSE: (sig(xy)-txy)^2 + (wh_logit - log(twh/anchor+1e-16))^2
    const float dx = sx - tx, dy = sy - ty;
    const float dw = p3 - logf(tw / aw + 1e-16f);
    const float dh = p4 - logf(th / ah + 1e-16f);
    acc_box = dx * dx + dy * dy + dw * dw + dh * dh;

    // class CE over 80 logits (only ~2% of cells reach here)
    const int cls = (int)tgt[(size_t)cell * TCH + 5];
    float m = -1e30f;
    for (int c = 0; c < NCLS; ++c) m = fmaxf(m, my[5 + c]);
    float se = 0.0f;
    for (int c = 0; c < NCLS; ++c) se += expf(my[5 + c] - m);
    acc_cls = (logf(se) + m) - my[5 + cls];
  }

  // ---- per-wave reduction via f32 WMMA (EXEC is all ones; exact grid) ----
  wave_reduce_pair_atomic(acc_noobj, acc_obj, lane, &ws[0], &ws[1]);
  wave_reduce_pair_atomic(acc_box, acc_cls, lane, &ws[2], &ws[3]);
  wave_reduce_pair_atomic(cnt_obj, cnt_noobj, lane, &ws[4], &ws[5]);
}

__global__ void yolo_final(const float* __restrict__ ws, float* __restrict__ out) {
  const float n_obj   = fmaxf(ws[4], 1.0f);
  const float n_noobj = fmaxf(ws[5], 1.0f);
  const float no_object_loss = ws[0] / n_noobj;
  const float object_loss    = ws[1] / n_obj;
  const float box_loss       = ws[2] / (n_obj * 4.0f);
  const float class_loss     = ws[3] / n_obj;
  out[0] = 10.0f * box_loss + 1.0f * object_loss + 1.0f * class_loss +
           10.0f * no_object_loss;
}

// ---------------- launch ----------------
extern "C" void kernel_launch(void* const* d_in, const int* in_sizes, int n_in,
                              void* d_out, int out_size, void* d_ws,
                              size_t ws_size, hipStream_t stream) {
  (void)in_sizes; (void)n_in; (void)out_size; (void)ws_size;
  const float* pred    = (const float*)d_in[0];
  const float* tgt     = (const float*)d_in[1];
  const float* anchors = (const float*)d_in[2];
  float* ws  = (float*)d_ws;
  float* out = (float*)d_out;

  yolo_zero_ws<<<1, 32, 0, stream>>>(ws);
  yolo_main<<<NBLOCKS, BDIM, 0, stream>>>(pred, tgt, anchors, ws);
  yolo_final<<<1, 1, 0, stream>>>(ws, out);
}